// GCN_14602888806887
// MI455X (gfx1250) — compile-verified
//
#include <hip/hip_runtime.h>

typedef float v2f __attribute__((ext_vector_type(2)));
typedef float v8f __attribute__((ext_vector_type(8)));

#define N_NODES 50000
#define F_IN    256
#define F_HID   128
#define F_OUT   40

// ---------------------------------------------------------------- degree
__global__ __launch_bounds__(256) void k_init_deg(float* __restrict__ deg) {
    int i = blockIdx.x * 256 + threadIdx.x;
    if (i < N_NODES) deg[i] = 1.0f;            // self-loop weight
}

__global__ __launch_bounds__(256) void k_deg(const int* __restrict__ col,
                                             const float* __restrict__ ew,
                                             float* __restrict__ deg, int E) {
    int e = blockIdx.x * 256 + threadIdx.x;
    if (e < E) atomicAdd(&deg[col[e]], ew[e]);
}

__global__ __launch_bounds__(256) void k_dinv(float* __restrict__ deg) {
    int i = blockIdx.x * 256 + threadIdx.x;
    if (i < N_NODES) {
        float d = deg[i];
        deg[i] = (d > 0.0f) ? rsqrtf(d) : 0.0f;   // in place: deg -> dinv
    }
}

// ---------------------------------------------------------------- GEMM1: h1 = X @ W1
// One wave32 computes a 16x64 strip: one A fragment feeds 4 WMMAs (A reuse x4).
// fp32 WMMA (V_WMMA_F32_16X16X4_F32), K stepped by 4.
__global__ __launch_bounds__(32) void k_gemm1(const float* __restrict__ X,
                                              const float* __restrict__ W,
                                              float* __restrict__ H) {
    const int m0   = blockIdx.x * 16;
    const int n0   = blockIdx.y * 64;      // 4 adjacent 16-wide N tiles
    const int lane = threadIdx.x;          // 0..31 (wave32)
    const int half = lane >> 4;            // 0: K pair {0,1}, 1: K pair {2,3}
    const int l16  = lane & 15;

    const float* __restrict__ arow = X + (long)(m0 + l16) * F_IN;
    const float* __restrict__ bcol = W + (n0 + l16);

    v8f acc0 = {}, acc1 = {}, acc2 = {}, acc3 = {};
#pragma unroll 4
    for (int k0 = 0; k0 < F_IN; k0 += 4) {
        const int kk = k0 + 2 * half;
        v2f a;
        a.x = arow[kk + 0];
        a.y = arow[kk + 1];
        const float* __restrict__ b0p = bcol + (long)kk * F_HID;
        const float* __restrict__ b1p = b0p + F_HID;
        v2f b;
        b.x = b0p[0];  b.y = b1p[0];
        acc0 = __builtin_amdgcn_wmma_f32_16x16x4_f32(false, a, false, b, (short)0, acc0, false, false);
        b.x = b0p[16]; b.y = b1p[16];
        acc1 = __builtin_amdgcn_wmma_f32_16x16x4_f32(false, a, false, b, (short)0, acc1, false, false);
        b.x = b0p[32]; b.y = b1p[32];
        acc2 = __builtin_amdgcn_wmma_f32_16x16x4_f32(false, a, false, b, (short)0, acc2, false, false);
        b.x = b0p[48]; b.y = b1p[48];
        acc3 = __builtin_amdgcn_wmma_f32_16x16x4_f32(false, a, false, b, (short)0, acc3, false, false);
    }
#pragma unroll
    for (int r = 0; r < 8; ++r) {
        int mm = m0 + r + 8 * half;        // C/D layout: row = r + 8*half, col = l16
        float* hrow = H + (long)mm * F_HID + n0 + l16;
        hrow[0]  = acc0[r];
        hrow[16] = acc1[r];
        hrow[32] = acc2[r];
        hrow[48] = acc3[r];
    }
}

// ---------------------------------------------------------------- self-loop seed layer 1
__global__ __launch_bounds__(256) void k_self1(const float* __restrict__ h1,
                                               const float* __restrict__ dinv,
                                               float* __restrict__ agg1) {
    long idx = (long)blockIdx.x * 256 + threadIdx.x;
    if (idx < (long)N_NODES * F_HID) {
        int  i  = (int)(idx >> 7);             // /128
        float di = dinv[i];
        agg1[idx] = h1[idx] * di * di;
    }
}

// ---------------------------------------------------------------- edge scatter layer 1
// One wave per edge; each lane moves a float4 (128 features / 32 lanes).
__global__ __launch_bounds__(256) void k_scatter1(const int* __restrict__ row,
                                                  const int* __restrict__ col,
                                                  const float* __restrict__ ew,
                                                  const float* __restrict__ dinv,
                                                  const float* __restrict__ h1,
                                                  float* __restrict__ agg1, int E) {
    int e    = blockIdx.x * 8 + (threadIdx.x >> 5);
    int lane = threadIdx.x & 31;
    if (e >= E) return;
    int r = row[e], c = col[e];
    float nrm = dinv[r] * ew[e] * dinv[c];
    const float4* __restrict__ src = (const float4*)(h1 + (long)r * F_HID);
    float4 v = src[lane];
    float* dst = agg1 + (long)c * F_HID + lane * 4;
    atomicAdd(dst + 0, v.x * nrm);
    atomicAdd(dst + 1, v.y * nrm);
    atomicAdd(dst + 2, v.z * nrm);
    atomicAdd(dst + 3, v.w * nrm);
}

// ---------------------------------------------------------------- bias + ReLU (in place)
__global__ __launch_bounds__(256) void k_relu(float* __restrict__ agg1,
                                              const float* __restrict__ b1) {
    long idx = (long)blockIdx.x * 256 + threadIdx.x;
    if (idx < (long)N_NODES * F_HID) {
        int j = (int)(idx & (F_HID - 1));
        float v = agg1[idx] + b1[j];
        agg1[idx] = v > 0.0f ? v : 0.0f;
    }
}

// ---------------------------------------------------------------- GEMM2: h2 = Hr @ W2
// One wave32 computes the full 16x48 strip (40 valid cols): A loaded exactly once.
__global__ __launch_bounds__(32) void k_gemm2(const float* __restrict__ Hr,
                                              const float* __restrict__ W,
                                              float* __restrict__ O) {
    const int m0   = blockIdx.x * 16;
    const int lane = threadIdx.x;
    const int half = lane >> 4;
    const int l16  = lane & 15;

    const float* __restrict__ arow = Hr + (long)(m0 + l16) * F_HID;
    const bool n2ok = (32 + l16) < F_OUT;      // third tile: cols 32..47, valid < 40

    v8f acc0 = {}, acc1 = {}, acc2 = {};
#pragma unroll 4
    for (int k0 = 0; k0 < F_HID; k0 += 4) {
        const int kk = k0 + 2 * half;
        v2f a;
        a.x = arow[kk + 0];
        a.y = arow[kk + 1];
        const float* __restrict__ b0p = W + (long)kk * F_OUT + l16;
        const float* __restrict__ b1p = b0p + F_OUT;
        v2f b;
        b.x = b0p[0];  b.y = b1p[0];
        acc0 = __builtin_amdgcn_wmma_f32_16x16x4_f32(false, a, false, b, (short)0, acc0, false, false);
        b.x = b0p[16]; b.y = b1p[16];
        acc1 = __builtin_amdgcn_wmma_f32_16x16x4_f32(false, a, false, b, (short)0, acc1, false, false);
        b.x = n2ok ? b0p[32] : 0.0f;
        b.y = n2ok ? b1p[32] : 0.0f;
        acc2 = __builtin_amdgcn_wmma_f32_16x16x4_f32(false, a, false, b, (short)0, acc2, false, false);
    }
#pragma unroll
    for (int r = 0; r < 8; ++r) {
        int mm = m0 + r + 8 * half;
        float* orow = O + (long)mm * F_OUT + l16;
        orow[0]  = acc0[r];
        orow[16] = acc1[r];
        if (n2ok) orow[32] = acc2[r];
    }
}

// ---------------------------------------------------------------- out = h2*dinv^2 + b2 (self-loop + bias)
__global__ __launch_bounds__(256) void k_outinit(const float* __restrict__ h2,
                                                 const float* __restrict__ dinv,
                                                 const float* __restrict__ b2,
                                                 float* __restrict__ out) {
    long idx = (long)blockIdx.x * 256 + threadIdx.x;
    if (idx < (long)N_NODES * F_OUT) {
        int i = (int)(idx / F_OUT);
        int j = (int)(idx - (long)i * F_OUT);
        float di = dinv[i];
        out[idx] = h2[idx] * di * di + b2[j];
    }
}

// ---------------------------------------------------------------- edge scatter layer 2 (40 features)
__global__ __launch_bounds__(256) void k_scatter2(const int* __restrict__ row,
                                                  const int* __restrict__ col,
                                                  const float* __restrict__ ew,
                                                  const float* __restrict__ dinv,
                                                  const float* __restrict__ h2,
                                                  float* __restrict__ out, int E) {
    int e    = blockIdx.x * 8 + (threadIdx.x >> 5);
    int lane = threadIdx.x & 31;
    if (e >= E) return;
    int r = row[e], c = col[e];
    float nrm = dinv[r] * ew[e] * dinv[c];
    const float* __restrict__ src = h2 + (long)r * F_OUT;
    float* dst = out + (long)c * F_OUT;
    atomicAdd(&dst[lane], src[lane] * nrm);
    if (lane < F_OUT - 32)
        atomicAdd(&dst[lane + 32], src[lane + 32] * nrm);
}

// ----------------------------------------------------------------
extern "C" void kernel_launch(void* const* d_in, const int* in_sizes, int n_in,
                              void* d_out, int out_size, void* d_ws, size_t ws_size,
                              hipStream_t stream) {
    const float* x   = (const float*)d_in[0];
    const int*   ei  = (const int*)d_in[1];
    const float* ew  = (const float*)d_in[2];
    const float* W1  = (const float*)d_in[3];
    const float* b1  = (const float*)d_in[4];
    const float* W2  = (const float*)d_in[5];
    const float* b2  = (const float*)d_in[6];
    float*       out = (float*)d_out;

    const int E = in_sizes[2];             // edge count (edge_weight length)
    const int* row = ei;
    const int* col = ei + E;

    // workspace layout (bytes, 512-aligned)
    char* ws = (char*)d_ws;
    float* deg  = (float*)(ws + 0);                         // N floats (becomes dinv)
    float* h1   = (float*)(ws + 409600);                    // N*128 floats
    float* agg1 = (float*)(ws + 409600 + 25600000);         // N*128 floats
    float* h2   = (float*)(ws + 409600 + 2 * 25600000);     // N*40 floats
    (void)ws_size; (void)n_in; (void)out_size;

    const int nb_nodes = (N_NODES + 255) / 256;
    const int nb_edges = (E + 255) / 256;
    const int nb_wave_edges = (E + 7) / 8;                  // 8 waves/block, 1 edge/wave
    const int nb_nh = (N_NODES * F_HID + 255) / 256;
    const int nb_no = (N_NODES * F_OUT + 255) / 256;

    // 1. normalization coefficients
    k_init_deg<<<nb_nodes, 256, 0, stream>>>(deg);
    k_deg<<<nb_edges, 256, 0, stream>>>(col, ew, deg, E);
    k_dinv<<<nb_nodes, 256, 0, stream>>>(deg);              // deg -> dinv in place
    float* dinv = deg;

    // 2. layer 1: h1 = x @ W1 (WMMA fp32, 16x64 strip per wave), aggregate, bias+relu
    k_gemm1<<<dim3(N_NODES / 16, F_HID / 64), 32, 0, stream>>>(x, W1, h1);
    k_self1<<<nb_nh, 256, 0, stream>>>(h1, dinv, agg1);
    k_scatter1<<<nb_wave_edges, 256, 0, stream>>>(row, col, ew, dinv, h1, agg1, E);
    k_relu<<<nb_nh, 256, 0, stream>>>(agg1, b1);

    // 3. layer 2: h2 = relu(h) @ W2 (WMMA fp32, full 16x48 strip per wave), aggregate into d_out
    k_gemm2<<<N_NODES / 16, 32, 0, stream>>>(agg1, W2, h2);
    k_outinit<<<nb_no, 256, 0, stream>>>(h2, dinv, b2, out);
    k_scatter2<<<nb_wave_edges, 256, 0, stream>>>(row, col, ew, dinv, h2, out, E);
}